// SEBlock_42906723287671
// MI455X (gfx1250) — compile-verified
//
#include <hip/hip_runtime.h>
#include <hip/hip_bf16.h>

// SE block, B=64 C=256 H=W=56, fp32.
// Bandwidth-bound: ~411MB min traffic -> ~18us floor at 23.3 TB/s.
// 3 kernels: squeeze (pass 1 over x), excite (1 WGP, WMMA f32 16x16x4), scale (pass 2, streaming).

#define Bsz 64
#define Csz 256
#define HID 64
#define HW  3136      // 56*56
#define HW4 784       // HW/4
#define LDSTRIDE 66   // 64 + 2 pad (keeps float2 alignment, spreads LDS banks)

typedef float v2f __attribute__((ext_vector_type(2)));
typedef float v4f __attribute__((ext_vector_type(4)));
typedef float v8f __attribute__((ext_vector_type(8)));

// ---------------- Kernel 1: squeeze (global average pool) ----------------
// One block per (b,c) plane: 3136 floats = 784 float4, 256 threads.
__global__ __launch_bounds__(256) void se_squeeze(const float* __restrict__ x,
                                                  float* __restrict__ y) {
    __shared__ float red[256];
    const int bc = blockIdx.x;
    const v4f* x4 = reinterpret_cast<const v4f*>(x + (size_t)bc * HW);
    float s = 0.0f;
    for (int i = threadIdx.x; i < HW4; i += 256) {
        v4f v = x4[i];
        s += (v.x + v.y) + (v.z + v.w);
    }
    red[threadIdx.x] = s;
    __syncthreads();
    for (int off = 128; off > 0; off >>= 1) {
        if (threadIdx.x < (unsigned)off) red[threadIdx.x] += red[threadIdx.x + off];
        __syncthreads();
    }
    if (threadIdx.x == 0) y[bc] = red[0] * (1.0f / (float)HW);
}

// ---------------- Kernel 2: excite (two tiny GEMMs via V_WMMA_F32_16X16X4_F32) ----
// Single workgroup, 512 threads = 16 wave32s.
// Phase A: H(64x64) = relu(Y(64x256) * W1^T(256x64) + b1)   -> LDS
// Phase B: G(64x256) = sigmoid(H(64x64) * W2^T(64x256) + b2) -> global ws
//
// WMMA f32 16x16x4 operand layout (wave32):
//   A (16x4):  lane L holds A[M=L&15][K = 2*(L>=16) + {0,1}] in {v0,v1}  -> float2 load
//   B (4x16):  lane L holds B[K = 2*(L>=16) + {0,1}][N=L&15]             -> float2 load of W^T
//   C/D:       VGPR i holds D[M = i + 8*(L>=16)][N = L&15]
__global__ __launch_bounds__(512) void se_excite(const float* __restrict__ y,
                                                 const float* __restrict__ w1,
                                                 const float* __restrict__ b1,
                                                 const float* __restrict__ w2,
                                                 const float* __restrict__ b2,
                                                 float* __restrict__ g) {
    __shared__ float Hs[HID * LDSTRIDE];

    const int tid  = threadIdx.x;
    const int wave = tid >> 5;
    const int lane = tid & 31;
    const int l15  = lane & 15;
    const int kb   = (lane >> 4) << 1;          // 0 or 2: K sub-offset for this lane half
    const int half = (lane >> 4) << 3;          // 0 or 8: M offset for C/D rows

    // ---- Phase A: 4x4 tile grid over 64x64, one tile per wave, K=256 ----
    {
        const int mt = wave >> 2;
        const int nt = wave & 3;
        const float* arow = y  + (mt * 16 + l15) * Csz;   // Y row (M)
        const float* brow = w1 + (nt * 16 + l15) * Csz;   // W1 row (N), B[k][n]=W1[n][k]
        v8f acc = {};
        #pragma unroll 8
        for (int k0 = 0; k0 < Csz; k0 += 4) {
            v2f a = *(const v2f*)(arow + k0 + kb);
            v2f b = *(const v2f*)(brow + k0 + kb);
            acc = __builtin_amdgcn_wmma_f32_16x16x4_f32(
                false, a, false, b, (short)0, acc, false, false);
        }
        const float bias = b1[nt * 16 + l15];
        const int   rb   = mt * 16 + half;
        #pragma unroll
        for (int i = 0; i < 8; ++i) {
            float v = acc[i] + bias;
            Hs[(rb + i) * LDSTRIDE + nt * 16 + l15] = v > 0.0f ? v : 0.0f;
        }
    }
    __syncthreads();

    // ---- Phase B: 4x16 tile grid over 64x256, four tiles per wave, K=64 ----
    {
        const int mt  = wave >> 2;
        const int ntb = wave & 3;
        const float* arow = Hs + (mt * 16 + l15) * LDSTRIDE;   // H row from LDS
        #pragma unroll
        for (int j = 0; j < 4; ++j) {
            const int nt = ntb + 4 * j;
            const float* brow = w2 + (nt * 16 + l15) * HID;    // W2 row (N), B[k][n]=W2[n][k]
            v8f acc = {};
            #pragma unroll
            for (int k0 = 0; k0 < HID; k0 += 4) {
                v2f a = *(const v2f*)(arow + k0 + kb);
                v2f b = *(const v2f*)(brow + k0 + kb);
                acc = __builtin_amdgcn_wmma_f32_16x16x4_f32(
                    false, a, false, b, (short)0, acc, false, false);
            }
            const float bias = b2[nt * 16 + l15];
            const int   rb   = mt * 16 + half;
            #pragma unroll
            for (int i = 0; i < 8; ++i) {
                float v = acc[i] + bias;
                v = 1.0f / (1.0f + __expf(-v));
                g[(rb + i) * Csz + nt * 16 + l15] = v;
            }
        }
    }
}

// ---------------- Kernel 3: scale (out = x * g broadcast) ----------------
// One block per (b,c) plane; g is block-uniform (scalarized). Streaming pass:
// nontemporal on x (last use) and out (no reuse) to spare the 192MB L2.
__global__ __launch_bounds__(256) void se_scale(const float* __restrict__ x,
                                                const float* __restrict__ g,
                                                float* __restrict__ out) {
    const int bc = blockIdx.x;
    const float gv = g[bc];
    const v4f* x4 = reinterpret_cast<const v4f*>(x + (size_t)bc * HW);
    v4f*       o4 = reinterpret_cast<v4f*>(out + (size_t)bc * HW);
    for (int i = threadIdx.x; i < HW4; i += 256) {
        v4f v = __builtin_nontemporal_load(&x4[i]);
        v *= gv;
        __builtin_nontemporal_store(v, &o4[i]);
    }
}

extern "C" void kernel_launch(void* const* d_in, const int* in_sizes, int n_in,
                              void* d_out, int out_size, void* d_ws, size_t ws_size,
                              hipStream_t stream) {
    const float* x  = (const float*)d_in[0];   // (64,256,56,56)
    const float* w1 = (const float*)d_in[1];   // (64,256)
    const float* b1 = (const float*)d_in[2];   // (64,)
    const float* w2 = (const float*)d_in[3];   // (256,64)
    const float* b2 = (const float*)d_in[4];   // (256,)
    float* out = (float*)d_out;

    float* y = (float*)d_ws;          // 64*256 floats
    float* g = y + Bsz * Csz;         // 64*256 floats (128KB total ws use)

    se_squeeze<<<Bsz * Csz, 256, 0, stream>>>(x, y);
    se_excite<<<1, 512, 0, stream>>>(y, w1, b1, w2, b2, g);
    se_scale<<<Bsz * Csz, 256, 0, stream>>>(x, g, out);
}